// Attention_8650064134594
// MI455X (gfx1250) — compile-verified
//
#include <hip/hip_runtime.h>
#include <hip/hip_bf16.h>

typedef float v2f __attribute__((ext_vector_type(2)));
typedef float v8f __attribute__((ext_vector_type(8)));

#define SEQ 8192
#define HID 4096
#define ROWS_PER_BLOCK 16
#define WAVES_PER_BLOCK 8
#define THREADS_PER_BLOCK (WAVES_PER_BLOCK * 32)
#define K_PER_WAVE (HID / WAVES_PER_BLOCK) // 512

// ws[0] = running max (as float bits), ws[1] = sum of exp
__global__ void att_init_kernel(float* ws) {
    if (threadIdx.x == 0) {
        ((unsigned int*)ws)[0] = 0xFF800000u; // -inf
        ws[1] = 0.0f;
    }
}

// GEMV via V_WMMA_F32_16X16X4_F32.
// Block: 256 threads = 8 waves. Block owns 16 rows; wave w owns K-slice
// [w*512, (w+1)*512). A-tile: 16 rows x 4 K per WMMA (lanes 0-15: K=0,1;
// lanes 16-31: K=2,3 -- ISA 32-bit A 16x4 layout). B: hidden broadcast to all
// 16 columns, so every column of D is the row-dot vector.
// Unroll 8 => 16 WMMAs + 32 b64 loads per body: ~4KB of A-bytes in flight per
// wave; x4096 waves ~= 16MB outstanding, enough to cover HBM latency at
// 23.3 TB/s.
__global__ __launch_bounds__(THREADS_PER_BLOCK) void att_gemv_wmma_kernel(
    const float* __restrict__ hidden, const float* __restrict__ eo,
    float* __restrict__ energies, float* __restrict__ ws) {
    __shared__ float lsum[ROWS_PER_BLOCK];

    const int tid = threadIdx.x;
    const int wave = tid >> 5;
    const int lane = tid & 31;
    const int rowbase = blockIdx.x * ROWS_PER_BLOCK;
    const int row = rowbase + (lane & 15);
    const int khalf = (lane >> 4) << 1; // 0 for lanes 0-15, 2 for lanes 16-31
    const int kbase = wave * K_PER_WAVE;

    if (tid < ROWS_PER_BLOCK) lsum[tid] = 0.0f;
    __syncthreads();

    const float* arow = eo + (size_t)row * HID + khalf; // 8B aligned
    const float* hrow = hidden + khalf;                 // 8B aligned

    v8f acc0 = {};
    v8f acc1 = {};
    #pragma unroll 8
    for (int kk = kbase; kk < kbase + K_PER_WAVE; kk += 8) {
        v2f a0 = *(const v2f*)(arow + kk);
        v2f b0 = *(const v2f*)(hrow + kk);
        v2f a1 = *(const v2f*)(arow + kk + 4);
        v2f b1 = *(const v2f*)(hrow + kk + 4);
        acc0 = __builtin_amdgcn_wmma_f32_16x16x4_f32(
            false, a0, false, b0, (short)0, acc0, false, false);
        acc1 = __builtin_amdgcn_wmma_f32_16x16x4_f32(
            false, a1, false, b1, (short)0, acc1, false, false);
    }

    // C/D layout: VGPR r holds M=r on lanes 0-15 and M=8+r on lanes 16-31;
    // all columns are identical, so lane 0 has rows 0..7, lane 16 rows 8..15.
    if (lane == 0) {
        #pragma unroll
        for (int i = 0; i < 8; ++i) atomicAdd(&lsum[i], acc0[i] + acc1[i]);
    } else if (lane == 16) {
        #pragma unroll
        for (int i = 0; i < 8; ++i) atomicAdd(&lsum[8 + i], acc0[i] + acc1[i]);
    }
    __syncthreads();

    if (tid < ROWS_PER_BLOCK) {
        float e = lsum[tid];
        energies[rowbase + tid] = e;
        // Global float max via signed/unsigned int atomics (handles negatives;
        // slot initialized to -inf bits).
        if (e >= 0.0f)
            atomicMax((int*)ws, __float_as_int(e));
        else
            atomicMin((unsigned int*)ws, __float_as_uint(e));
    }
}

__global__ __launch_bounds__(256) void att_exp_sum_kernel(float* out, float* ws) {
    __shared__ float ssum[256];
    const int i = blockIdx.x * 256 + threadIdx.x;
    const float m = ws[0];
    const float p = expf(out[i] - m);
    out[i] = p;
    ssum[threadIdx.x] = p;
    __syncthreads();
    #pragma unroll
    for (int s = 128; s > 0; s >>= 1) {
        if (threadIdx.x < s) ssum[threadIdx.x] += ssum[threadIdx.x + s];
        __syncthreads();
    }
    if (threadIdx.x == 0) atomicAdd(ws + 1, ssum[0]);
}

__global__ __launch_bounds__(256) void att_norm_kernel(float* out, const float* ws) {
    const int i = blockIdx.x * 256 + threadIdx.x;
    out[i] = out[i] / ws[1];
}

extern "C" void kernel_launch(void* const* d_in, const int* in_sizes, int n_in,
                              void* d_out, int out_size, void* d_ws, size_t ws_size,
                              hipStream_t stream) {
    const float* hidden = (const float*)d_in[0]; // [1, 4096]
    const float* eo     = (const float*)d_in[1]; // [8192, 4096]
    float* out = (float*)d_out;                  // [8192] energies -> attn
    float* ws  = (float*)d_ws;                   // ws[0]=max, ws[1]=sum

    att_init_kernel<<<1, 32, 0, stream>>>(ws);
    att_gemv_wmma_kernel<<<SEQ / ROWS_PER_BLOCK, THREADS_PER_BLOCK, 0, stream>>>(hidden, eo, out, ws);
    att_exp_sum_kernel<<<SEQ / 256, 256, 0, stream>>>(out, ws);
    att_norm_kernel<<<SEQ / 256, 256, 0, stream>>>(out, ws);
}